// LSTM4_54528904790334
// MI455X (gfx1250) — compile-verified
//
#include <hip/hip_runtime.h>

// CDNA5 / gfx1250 fused LSTM (B=16384, T=64, IN=78, H=8) + MLP(8->16->11) + softmax.
// One wave (32 lanes) processes 16 batch rows using fp32 WMMA 16x16x4 in the
// transposed formulation gates^T = W * [x_t | h]^T.

typedef __attribute__((ext_vector_type(2))) float v2f;
typedef __attribute__((ext_vector_type(8))) float v8f;

#define WMMA4(A, B, C) \
  __builtin_amdgcn_wmma_f32_16x16x4_f32(false, (A), false, (B), (short)0, (C), false, false)

__device__ __forceinline__ float fast_rcp(float x) { return __builtin_amdgcn_rcpf(x); }
__device__ __forceinline__ float fast_sigmoid(float x) { return fast_rcp(1.f + __expf(-x)); }
__device__ __forceinline__ float fast_tanh(float x) {
  return fmaf(2.f, fast_rcp(1.f + __expf(-2.f * x)), -1.f);
}

#define T_STEPS 64
#define IN_DIM  78
#define N_CLS   11

__global__ __launch_bounds__(32)
void lstm_fused_kernel(const float* __restrict__ x,
                       const float* __restrict__ Wih,
                       const float* __restrict__ Whh,
                       const float* __restrict__ bih,
                       const float* __restrict__ bhh,
                       const float* __restrict__ W1,
                       const float* __restrict__ b1p,
                       const float* __restrict__ W2,
                       const float* __restrict__ b2p,
                       float* __restrict__ out)
{
  const int lane = threadIdx.x & 31;
  const int c    = lane >> 4;   // lane group: K-half for A/B, gate-half for C/D
  const int r    = lane & 15;   // A row within tile / batch column for B,C,D
  const long bt  = (long)blockIdx.x * 16;

  // ---- Pack W_ih into fp32 WMMA A layout; persistent in VGPRs across all t ----
  // A tile0 = gate rows 0..15 (i|f), tile1 = gate rows 16..31 (g|o).
  // Lane(c,r), chunk k: v0 = W[row, 4k+2c], v1 = W[row, 4k+2c+1].
  v2f A0[20], A1[20];
#pragma unroll
  for (int k = 0; k < 19; ++k) {
    A0[k] = *(const v2f*)(Wih + (r)      * IN_DIM + 4 * k + 2 * c);
    A1[k] = *(const v2f*)(Wih + (16 + r) * IN_DIM + 4 * k + 2 * c);
  }
  { // chunk 19: K = 76,77 valid only in group0 (78 padded to 80)
    v2f a0; a0[0] = 0.f; a0[1] = 0.f;
    v2f a1 = a0;
    if (c == 0) {
      a0 = *(const v2f*)(Wih + r * IN_DIM + 76);
      a1 = *(const v2f*)(Wih + (16 + r) * IN_DIM + 76);
    }
    A0[19] = a0; A1[19] = a1;
  }
  // W_hh (32x8): 2 gate tiles x 2 K-chunks
  v2f Ah0[2], Ah1[2];
#pragma unroll
  for (int k = 0; k < 2; ++k) {
    Ah0[k] = *(const v2f*)(Whh + r * 8        + 4 * k + 2 * c);
    Ah1[k] = *(const v2f*)(Whh + (16 + r) * 8 + 4 * k + 2 * c);
  }
  // Bias broadcast in C/D layout: acc0 VGPR v <-> gate 8c+v, acc1 <-> gate 16+8c+v.
  float bias0[8], bias1[8];
#pragma unroll
  for (int v = 0; v < 8; ++v) {
    const int g0 = 8 * c + v, g1 = 16 + 8 * c + v;
    bias0[v] = bih[g0] + bhh[g0];
    bias1[v] = bih[g1] + bhh[g1];
  }

  // LSTM state for batch (bt+r), owned by group-0 lanes.
  float h[8], cstate[8];
#pragma unroll
  for (int v = 0; v < 8; ++v) { h[v] = 0.f; cstate[v] = 0.f; }

  // B-matrix source: lane(c,b) reads x[bt+b, t, 4k+2c .. 4k+2c+1] (8B aligned).
  const float* xrow = x + (bt + r) * (long)(T_STEPS * IN_DIM) + 2 * c;

  const float sc = c ? 1.f : 2.f;   // s1: tanh in group0, sigmoid in group1
  const float mo = c ? 0.f : -1.f;

  for (int t = 0; t < T_STEPS; ++t) {
    const float* xt = xrow + t * IN_DIM;

    v8f acc0, acc1;   // gates^T tiles: [gate 8c+v, batch r] / [gate 16+8c+v, batch r]
#pragma unroll
    for (int v = 0; v < 8; ++v) { acc0[v] = bias0[v]; acc1[v] = bias1[v]; }

    // Input projection: 20 K-chunks of 4 over the 78-wide row (padded with 0).
#pragma unroll
    for (int k = 0; k < 19; ++k) {
      v2f bx = *(const v2f*)(xt + 4 * k);
      acc0 = WMMA4(A0[k], bx, acc0);
      acc1 = WMMA4(A1[k], bx, acc1);
    }
    {
      v2f bx; bx[0] = 0.f; bx[1] = 0.f;
      if (c == 0) bx = *(const v2f*)(xt + 76);   // K=76,77; group1 half is padding
      acc0 = WMMA4(A0[19], bx, acc0);
      acc1 = WMMA4(A1[19], bx, acc1);
    }

    // Recurrence: gates^T += W_hh * h^T. B layout needs group1 to carry h2,h3 / h6,h7.
    const float h2s = __shfl_xor(h[2], 16, 32);
    const float h3s = __shfl_xor(h[3], 16, 32);
    const float h6s = __shfl_xor(h[6], 16, 32);
    const float h7s = __shfl_xor(h[7], 16, 32);
    v2f hb0, hb1;
    hb0[0] = c ? h2s : h[0]; hb0[1] = c ? h3s : h[1];
    hb1[0] = c ? h6s : h[4]; hb1[1] = c ? h7s : h[5];
    acc0 = WMMA4(Ah0[0], hb0, acc0);
    acc1 = WMMA4(Ah1[0], hb0, acc1);
    acc0 = WMMA4(Ah0[1], hb1, acc0);
    acc1 = WMMA4(Ah1[1], hb1, acc1);

    // Nonlinearities: s0 = sigmoid(acc0) -> i(c=0)/f(c=1); s1 -> tanh g(c=0)/sigmoid o(c=1).
    float s0[8], s1[8];
#pragma unroll
    for (int v = 0; v < 8; ++v) {
      s0[v] = fast_sigmoid(acc0[v]);
      s1[v] = fmaf(sc, fast_rcp(1.f + __expf(-sc * acc1[v])), mo);
    }
    // Cell update in group-0 lanes: f,o arrive from partner lane b+16.
#pragma unroll
    for (int v = 0; v < 8; ++v) {
      const float fv = __shfl_xor(s0[v], 16, 32);
      const float ov = __shfl_xor(s1[v], 16, 32);
      const float cn = fmaf(fv, cstate[v], s0[v] * s1[v]);  // f*c + i*g
      cstate[v] = cn;
      h[v] = ov * fast_tanh(cn);                            // o*tanh(c)
    }
  }

  // ---- Head: relu -> Linear(8,16) -> relu -> Linear(16,11) -> softmax ----
  if (c == 0) {
    float hr[8];
#pragma unroll
    for (int v = 0; v < 8; ++v) hr[v] = fmaxf(h[v], 0.f);

    float z1[16];
#pragma unroll
    for (int j = 0; j < 16; ++j) {
      float a = b1p[j];
#pragma unroll
      for (int v = 0; v < 8; ++v) a = fmaf(W1[j * 8 + v], hr[v], a);
      z1[j] = fmaxf(a, 0.f);
    }

    float z2[N_CLS];
    float m = -3.4e38f;
#pragma unroll
    for (int j = 0; j < N_CLS; ++j) {
      float a = b2p[j];
#pragma unroll
      for (int v = 0; v < 16; ++v) a = fmaf(W2[j * 16 + v], z1[v], a);
      z2[j] = a;
      m = fmaxf(m, a);
    }
    float s = 0.f;
#pragma unroll
    for (int j = 0; j < N_CLS; ++j) { z2[j] = __expf(z2[j] - m); s += z2[j]; }
    const float inv = fast_rcp(s);

    float* op = out + (bt + r) * N_CLS;
#pragma unroll
    for (int j = 0; j < N_CLS; ++j) op[j] = z2[j] * inv;
  }
}

extern "C" void kernel_launch(void* const* d_in, const int* in_sizes, int n_in,
                              void* d_out, int out_size, void* d_ws, size_t ws_size,
                              hipStream_t stream) {
  (void)n_in; (void)out_size; (void)d_ws; (void)ws_size;
  const float* x   = (const float*)d_in[0];
  const float* Wih = (const float*)d_in[1];
  const float* Whh = (const float*)d_in[2];
  const float* bih = (const float*)d_in[3];
  const float* bhh = (const float*)d_in[4];
  const float* W1  = (const float*)d_in[5];
  const float* b1  = (const float*)d_in[6];
  const float* W2  = (const float*)d_in[7];
  const float* b2  = (const float*)d_in[8];

  const int B = in_sizes[0] / (T_STEPS * IN_DIM);   // 16384
  dim3 grid(B / 16), block(32);
  lstm_fused_kernel<<<grid, block, 0, stream>>>(x, Wih, Whh, bih, bhh,
                                                W1, b1, W2, b2, (float*)d_out);
}